// SpGraphAttentionLayer_3229815406783
// MI455X (gfx1250) — compile-verified
//
#include <hip/hip_runtime.h>
#include <cstdint>
#include <cstddef>

typedef __attribute__((ext_vector_type(2))) float v2f;
typedef __attribute__((ext_vector_type(8))) float v8f;
typedef int v4i_ __attribute__((__vector_size__(16)));

#define D_FEAT 128
#define R_FEAT 64
#define ACOLS  320   // a is (128, 2*128+64)

#if __has_builtin(__builtin_amdgcn_global_load_async_to_lds_b128) && \
    __has_builtin(__builtin_amdgcn_s_wait_asynccnt)
#define HAVE_ASYNC_LDS 1
#else
#define HAVE_ASYNC_LDS 0
#endif

__device__ __forceinline__ void atomAddF32(float* p, float v) {
  __hip_atomic_fetch_add(p, v, __ATOMIC_RELAXED, __HIP_MEMORY_SCOPE_AGENT);
}

#if HAVE_ASYNC_LDS
// One 16-byte chunk, global -> LDS, per active lane (ASYNCcnt-tracked).
__device__ __forceinline__ void asyncCopy16(const float* g, float* l) {
  __builtin_amdgcn_global_load_async_to_lds_b128(
      (__attribute__((address_space(1))) v4i_*)(uintptr_t)g,
      (__attribute__((address_space(3))) v4i_*)l, 0, 0);
}
#endif

// u[j] = sum_o a_2[o] * a[o][j]   (j in [0,320))  -> u_src|u_dst|u_rel
__global__ void gat_compute_u(const float* __restrict__ aW,
                              const float* __restrict__ a2,
                              float* __restrict__ u) {
  int j = threadIdx.x;
  if (j < ACOLS) {
    float s = 0.f;
    for (int o = 0; o < D_FEAT; ++o) s += a2[o] * aW[o * ACOLS + j];
    u[j] = s;
  }
}

// Y_src = x @ a_src^T, Y_dst = x @ a_dst^T via f32 WMMA (16x16x4).
// Block = 8 waves; each wave does two 16x16 C tiles; combined 256 output cols.
__global__ __launch_bounds__(256) void gat_node_proj(
    const float* __restrict__ x, const float* __restrict__ aW,
    float* __restrict__ Ysrc, float* __restrict__ Ydst, int N) {
  const int wave = threadIdx.x >> 5;
  const int lane = threadIdx.x & 31;
  const int lrow = lane & 15;
  const int lhi  = lane >> 4;
  const int m0 = blockIdx.x * 16;
  int arow = m0 + lrow; if (arow >= N) arow = N - 1;
  const float* xr = x + (size_t)arow * D_FEAT;

  for (int ct = wave * 2; ct < wave * 2 + 2; ++ct) {
    const int n = ct * 16 + lrow;  // combined col 0..255
    const float* wrow = (n < D_FEAT) ? (aW + (size_t)n * ACOLS)
                                     : (aW + (size_t)(n - D_FEAT) * ACOLS + D_FEAT);
    v8f acc = {};
#pragma unroll
    for (int k0 = 0; k0 < D_FEAT; k0 += 4) {
      const int ka = k0 + 2 * lhi;
      v2f af, bf;
      af.x = xr[ka];   af.y = xr[ka + 1];
      bf.x = wrow[ka]; bf.y = wrow[ka + 1];
      acc = __builtin_amdgcn_wmma_f32_16x16x4_f32(false, af, false, bf,
                                                  (short)0, acc, false, false);
    }
#pragma unroll
    for (int v = 0; v < 8; ++v) {
      const int m = m0 + v + 8 * lhi;
      if (m < N) {
        if (n < D_FEAT) Ysrc[(size_t)m * D_FEAT + n] = acc[v];
        else            Ydst[(size_t)m * D_FEAT + (n - D_FEAT)] = acc[v];
      }
    }
  }
}

// s_src[n] = x[n].u_src ; s_dst[n] = x[n].u_dst
__global__ void gat_node_scores(const float* __restrict__ x,
                                const float* __restrict__ u,
                                float* __restrict__ ssrc,
                                float* __restrict__ sdst, int N) {
  int n = blockIdx.x * blockDim.x + threadIdx.x;
  if (n >= N) return;
  const float4* xr = (const float4*)(x + (size_t)n * D_FEAT);
  const float4* us = (const float4*)u;
  const float4* ud = (const float4*)(u + D_FEAT);
  float a = 0.f, b = 0.f;
#pragma unroll 8
  for (int i = 0; i < D_FEAT / 4; ++i) {
    float4 xv = xr[i], uv = us[i], vv = ud[i];
    a += xv.x * uv.x + xv.y * uv.y + xv.z * uv.z + xv.w * uv.w;
    b += xv.x * vv.x + xv.y * vv.y + xv.z * vv.z + xv.w * vv.w;
  }
  ssrc[n] = a; sdst[n] = b;
}

// Per-edge kernel: score -> edge_e -> rowsum atomics; R_e via WMMA; scatter
// ee*(Y_dst[dst]+R_e) into agg[src] (d_out used as accumulator).
// LDS panels filled with GLOBAL_LOAD_ASYNC_TO_LDS_B128 when available.
__global__ __launch_bounds__(128) void gat_edge_kernel(
    const int* __restrict__ edge1, const float* __restrict__ emb1,
    const int* __restrict__ edge2, const float* __restrict__ emb2,
    const float* __restrict__ aW, const float* __restrict__ u,
    const float* __restrict__ ssrc, const float* __restrict__ sdst,
    const float* __restrict__ Ydst,
    float* __restrict__ rowsum, float* __restrict__ agg,
    int E1, int E2) {
  const int E = E1 + E2;
  __shared__ float sBnk[D_FEAT][68];     // a_rel row-major [n][k], padded pitch
  __shared__ float sEmb[4][16][68];      // per-wave 16x64, padded pitch
  __shared__ float sEE[4][16];
  __shared__ int   sS[4][16];
  __shared__ int   sD[4][16];

  const int tid  = threadIdx.x;
  const int wave = tid >> 5;
  const int lane = tid & 31;
  const int lrow = lane & 15;
  const int lhi  = lane >> 4;
  const int eBase = blockIdx.x * 64 + wave * 16;

#if HAVE_ASYNC_LDS
  // B panel: 128 rows x 16 chunks of 16B, contiguous per chunk.
  for (int c = tid; c < D_FEAT * 16; c += 128) {
    const int n = c >> 4, cc = c & 15;
    asyncCopy16(aW + (size_t)n * ACOLS + 2 * D_FEAT + cc * 4, &sBnk[n][cc * 4]);
  }
  // Embed tile: per wave 16 rows x 16 chunks of 16B (64 floats/row).
  {
#pragma unroll
    for (int it = 0; it < 8; ++it) {
      const int c = lane + 32 * it;      // 0..255 chunks for this wave
      const int r = c >> 4, cc = c & 15;
      int e = eBase + r;
      int ec = e < E ? e : E - 1;
      const float* rowp = (ec < E1) ? (emb1 + (size_t)ec * R_FEAT)
                                    : (emb2 + (size_t)(ec - E1) * R_FEAT);
      if (it == 0) __builtin_prefetch(rowp + 64 * R_FEAT, 0, 1);
      asyncCopy16(rowp + cc * 4, &sEmb[wave][r][cc * 4]);
    }
  }
#else
  // Fallback: register-staged copies with identical LDS layout.
  for (int c = tid; c < D_FEAT * 16; c += 128) {
    const int n = c >> 4, cc = c & 15;
    const float4 vq = *(const float4*)(aW + (size_t)n * ACOLS + 2 * D_FEAT + cc * 4);
    sBnk[n][cc * 4 + 0] = vq.x; sBnk[n][cc * 4 + 1] = vq.y;
    sBnk[n][cc * 4 + 2] = vq.z; sBnk[n][cc * 4 + 3] = vq.w;
  }
  {
    const int r  = lane >> 1;
    const int c0 = (lane & 1) * 32;
    int e = eBase + r;
    int ec = e < E ? e : E - 1;
    const float* rowp = (ec < E1) ? (emb1 + (size_t)ec * R_FEAT)
                                  : (emb2 + (size_t)(ec - E1) * R_FEAT);
    __builtin_prefetch(rowp + 64 * R_FEAT, 0, 1);
    const float4* rp4 = (const float4*)(rowp + c0);
#pragma unroll
    for (int q = 0; q < 8; ++q) {
      float4 vq = rp4[q];
      sEmb[wave][r][c0 + q * 4 + 0] = vq.x;
      sEmb[wave][r][c0 + q * 4 + 1] = vq.y;
      sEmb[wave][r][c0 + q * 4 + 2] = vq.z;
      sEmb[wave][r][c0 + q * 4 + 3] = vq.w;
    }
  }
#endif

  if (lane < 16) {
    int e = eBase + lane;
    int ec = e < E ? e : E - 1;
    int s, d;
    if (ec < E1) { s = edge1[ec]; d = edge1[E1 + ec]; }
    else { int e2 = ec - E1; s = edge2[e2]; d = edge2[E2 + e2]; }
    sS[wave][lane] = s;
    sD[wave][lane] = d;
  }

#if HAVE_ASYNC_LDS
  __builtin_amdgcn_s_wait_asynccnt(0);
#endif
  __syncthreads();

  // scores: one edge per low lane
  if (lane < 16) {
    const int e = eBase + lane;
    const bool valid = e < E;
    const int s = sS[wave][lane], d = sD[wave][lane];
    float sc = ssrc[s] + sdst[d];
#pragma unroll 8
    for (int k = 0; k < R_FEAT; ++k) sc += sEmb[wave][lane][k] * u[2 * D_FEAT + k];
    const float ee = expf(-(sc > 0.f ? sc : 0.2f * sc));
    sEE[wave][lane] = valid ? ee : 0.f;
    if (valid) atomAddF32(&rowsum[s], ee);
  }
  __syncthreads();

  float eeR[8]; int sR[8]; int dR[8];
#pragma unroll
  for (int v = 0; v < 8; ++v) {
    const int m = v + 8 * lhi;
    eeR[v] = sEE[wave][m];
    sR[v]  = sS[wave][m];
    dR[v]  = sD[wave][m];
  }

  // R_e = emb(16x64) @ a_rel^T(64x128), one 16x16 n-tile at a time, fused scatter
#pragma unroll 1
  for (int t = 0; t < 8; ++t) {
    v8f acc = {};
    const int n = t * 16 + lrow;
#pragma unroll
    for (int k0 = 0; k0 < R_FEAT; k0 += 4) {
      const int ka = k0 + 2 * lhi;
      v2f af, bf;
      af.x = sEmb[wave][lrow][ka];
      af.y = sEmb[wave][lrow][ka + 1];
      bf.x = sBnk[n][ka];          // B[k][n] == a_rel[n][k]
      bf.y = sBnk[n][ka + 1];
      acc = __builtin_amdgcn_wmma_f32_16x16x4_f32(false, af, false, bf,
                                                  (short)0, acc, false, false);
    }
#pragma unroll
    for (int v = 0; v < 8; ++v) {
      const int e = eBase + v + 8 * lhi;
      if (e < E) {
        const float val = acc[v] + Ydst[(size_t)dR[v] * D_FEAT + n];
        atomAddF32(&agg[(size_t)sR[v] * D_FEAT + n], eeR[v] * val);
      }
    }
  }
}

// h = (rowsum*Y_src + agg) / (rowsum==0 ? 1e-12 : rowsum); out = elu(h)
__global__ void gat_finalize(const float* __restrict__ Ysrc,
                             const float* __restrict__ rowsum,
                             float* __restrict__ out, int total) {
  int i = blockIdx.x * blockDim.x + threadIdx.x;
  if (i >= total) return;
  const int n = i >> 7;
  const float rs = rowsum[n];
  const float num = rs * Ysrc[i] + out[i];
  const float h = num / (rs == 0.f ? 1e-12f : rs);
  out[i] = h > 0.f ? h : expm1f(h);
}

extern "C" void kernel_launch(void* const* d_in, const int* in_sizes, int n_in,
                              void* d_out, int out_size, void* d_ws, size_t ws_size,
                              hipStream_t stream) {
  const float* x     = (const float*)d_in[0];
  const int*   edge1 = (const int*)d_in[1];
  const float* emb1  = (const float*)d_in[2];
  const int*   edge2 = (const int*)d_in[3];
  const float* emb2  = (const float*)d_in[4];
  const float* aW    = (const float*)d_in[5];
  const float* a2    = (const float*)d_in[6];

  const int N  = in_sizes[0] / D_FEAT;
  const int E1 = in_sizes[1] / 2;
  const int E2 = in_sizes[3] / 2;
  const int E  = E1 + E2;

  float* out       = (float*)d_out;
  float* ws_u      = (float*)d_ws;                 // 320
  float* ws_ssrc   = ws_u + ACOLS;                 // N
  float* ws_sdst   = ws_ssrc + N;                  // N
  float* ws_rowsum = ws_sdst + N;                  // N
  float* ws_Ysrc   = ws_rowsum + N;                // N*128
  float* ws_Ydst   = ws_Ysrc + (size_t)N * D_FEAT; // N*128

  (void)hipMemsetAsync(d_out, 0, (size_t)out_size * sizeof(float), stream);
  (void)hipMemsetAsync(ws_rowsum, 0, (size_t)N * sizeof(float), stream);

  gat_compute_u<<<1, ACOLS, 0, stream>>>(aW, a2, ws_u);
  gat_node_proj<<<(N + 15) / 16, 256, 0, stream>>>(x, aW, ws_Ysrc, ws_Ydst, N);
  gat_node_scores<<<(N + 255) / 256, 256, 0, stream>>>(x, ws_u, ws_ssrc, ws_sdst, N);
  gat_edge_kernel<<<(E + 63) / 64, 128, 0, stream>>>(edge1, emb1, edge2, emb2,
                                                     aW, ws_u, ws_ssrc, ws_sdst,
                                                     ws_Ydst, ws_rowsum, out, E1, E2);
  const int total = N * D_FEAT;
  gat_finalize<<<(total + 255) / 256, 256, 0, stream>>>(ws_Ysrc, ws_rowsum, out, total);
}